// SparseSelector_87067577024703
// MI455X (gfx1250) — compile-verified
//
#include <hip/hip_runtime.h>
#include <hip/hip_bf16.h>
#include <stdint.h>

// Problem constants (from reference): B=32, N=4096, D=1024, k = 0.4*N = 1638
#define B_BATCH 32
#define N_TOK   4096
#define D_FEAT  1024
#define K_SEL   1638

// d_out layout (flat float, reference return order):
//   [0)                      selected_compact  B*K*D
//   [B*K*D)                  mask_expanded     B*N
//   [B*K*D + B*N)            indices (as f32)  B*K
#define OUT_SEL_OFF  ((size_t)0)
#define OUT_MASK_OFF ((size_t)B_BATCH * K_SEL * D_FEAT)
#define OUT_IDX_OFF  (OUT_MASK_OFF + (size_t)B_BATCH * N_TOK)

typedef uint32_t u32x4 __attribute__((ext_vector_type(4)));
typedef uint32_t u32x8 __attribute__((ext_vector_type(8)));

// ---------------------------------------------------------------------------
// Kernel 1: per-batch top-k via 64-bit-key bitonic sort in LDS.
// key = (~monotone(score)) << 32 | index  -> ascending u64 sort gives
// descending score with ascending-index tie-break (matches lax.top_k).
// ---------------------------------------------------------------------------
__global__ __launch_bounds__(1024) void topk_sort_kernel(
    const float* __restrict__ scores,      // [B, N, 1]
    float*       __restrict__ out_idx_f,   // [B, K] (float-encoded indices)
    float*       __restrict__ out_mask,    // [B, N]
    int*         __restrict__ ws_idx)      // [B, K] int indices for gather
{
    __shared__ unsigned long long key[N_TOK];   // 32 KB LDS

    const int b = blockIdx.x;
    const int t = threadIdx.x;

    for (int i = t; i < N_TOK; i += 1024) {
        float s = scores[(size_t)b * N_TOK + i];
        unsigned u = __float_as_uint(s);
        unsigned m = (u & 0x80000000u) ? ~u : (u | 0x80000000u); // ascending map
        unsigned d = ~m;                                          // descending
        key[i] = ((unsigned long long)d << 32) | (unsigned)i;
    }
    __syncthreads();

    for (int size = 2; size <= N_TOK; size <<= 1) {
        for (int stride = size >> 1; stride > 0; stride >>= 1) {
            for (int p = t; p < (N_TOK / 2); p += 1024) {
                int pos     = 2 * p - (p & (stride - 1));
                int partner = pos + stride;
                bool up = ((pos & size) == 0);
                unsigned long long a = key[pos];
                unsigned long long c = key[partner];
                if ((a > c) == up) { key[pos] = c; key[partner] = a; }
            }
            __syncthreads();
        }
    }

    for (int j = t; j < N_TOK; j += 1024) {
        int idx = (int)(key[j] & 0xFFFFFFFFull);
        out_mask[(size_t)b * N_TOK + idx] = (j < K_SEL) ? 1.0f : 0.0f;
        if (j < K_SEL) {
            out_idx_f[(size_t)b * K_SEL + j] = (float)idx;
            ws_idx[b * K_SEL + j]            = idx;
        }
    }
}

// ---------------------------------------------------------------------------
// Kernel 2: bulk gather via Tensor Data Mover (gather mode).
// One wave per block. tensor_load_to_lds gathers 16 rows (16-bit indices in
// D# groups 2/3) of 1024 f32 into LDS; tensor_store_from_lds writes the
// contiguous 16x1024 tile to the output. Tracked with TENSORcnt.
// Descriptor packing per ISA 8.3-8.7:
//   G0: dw0 = count=1 | gather_mode<<31 (load only); dw1 = lds_addr;
//       dw2 = gaddr[31:0]; dw3 = gaddr[56:32] | type(2)<<30
//   G1: dw0 = data_size(4B=2)<<16; dw1[31:16]=tensor_dim0 lo;
//       dw2[31:16]=tensor_dim1 lo; dw3[31:16]=tile_dim0; dw4=tile_dim1;
//       dw5=tensor_dim0_stride lo; dw6,dw7=0
//   G2/G3 (gather): packed 16-bit row indices 0..15
// ---------------------------------------------------------------------------
#define TDM_ROWS 16
#define TDM_GROUPS (K_SEL / TDM_ROWS)     // 102 full groups -> rows 0..1631
#define TDM_TAIL  (K_SEL - TDM_GROUPS * TDM_ROWS)  // 6 remainder rows

__global__ __launch_bounds__(32) void gather_tdm_kernel(
    const float* __restrict__ vf,      // [B, N, D]
    const int*   __restrict__ ws_idx,  // [B, K]
    float*       __restrict__ out_sel) // [B, K, D]
{
    extern __shared__ char smem[];     // 64 KB dynamic LDS staging

    const int b  = blockIdx.y;
    const int j0 = blockIdx.x * TDM_ROWS;

    // Pack 16 row indices as 16-bit pairs (uniform -> SGPRs via "s").
    uint32_t pk[8];
    #pragma unroll
    for (int r = 0; r < 8; ++r) {
        uint32_t lo = (uint32_t)ws_idx[b * K_SEL + j0 + 2 * r];
        uint32_t hi = (uint32_t)ws_idx[b * K_SEL + j0 + 2 * r + 1];
        pk[r] = (lo & 0xFFFFu) | (hi << 16);
    }

    uint64_t ga_in  = (uint64_t)(uintptr_t)(vf + (size_t)b * N_TOK * D_FEAT);
    uint64_t ga_out = (uint64_t)(uintptr_t)
                      (out_sel + ((size_t)b * K_SEL + j0) * D_FEAT);
    uint32_t lds    = (uint32_t)(uintptr_t)(void*)smem;

    // ---- gather load descriptor ----
    u32x4 L0 = { 0x80000001u,                       // count=1, gather, 16b idx
                 lds,
                 (uint32_t)ga_in,
                 (uint32_t)((ga_in >> 32) & 0x01FFFFFFu) | (2u << 30) };
    u32x8 L1 = { 0x00020000u,                       // data_size = 4B
                 (uint32_t)D_FEAT << 16,            // tensor_dim0 = 1024
                 (uint32_t)N_TOK  << 16,            // tensor_dim1 = 4096
                 (uint32_t)D_FEAT << 16,            // tile_dim0   = 1024
                 (uint32_t)TDM_ROWS,                // tile_dim1 = #indices
                 (uint32_t)D_FEAT,                  // tensor_dim0_stride
                 0u, 0u };
    u32x4 L2 = { pk[0], pk[1], pk[2], pk[3] };      // row_index 0..7
    u32x4 L3 = { pk[4], pk[5], pk[6], pk[7] };      // row_index 8..15

    asm volatile("tensor_load_to_lds %0, %1, %2, %3"
                 :: "s"(L0), "s"(L1), "s"(L2), "s"(L3)
                 : "memory");
    __builtin_amdgcn_s_wait_tensorcnt(0);           // LDS tile resident

    // ---- contiguous 2D store descriptor ----
    u32x4 S0 = { 0x00000001u,                       // count=1, normal mode
                 lds,
                 (uint32_t)ga_out,
                 (uint32_t)((ga_out >> 32) & 0x01FFFFFFu) | (2u << 30) };
    u32x8 S1 = { 0x00020000u,                       // data_size = 4B
                 (uint32_t)D_FEAT << 16,            // tensor_dim0 = 1024
                 (uint32_t)TDM_ROWS << 16,          // tensor_dim1 = 16
                 (uint32_t)D_FEAT << 16,            // tile_dim0   = 1024
                 (uint32_t)TDM_ROWS,                // tile_dim1   = 16
                 (uint32_t)D_FEAT,                  // tensor_dim0_stride
                 0u, 0u };
    u32x4 Z  = { 0u, 0u, 0u, 0u };                  // dims 3/4 unused

    asm volatile("tensor_store_from_lds %0, %1, %2, %3"
                 :: "s"(S0), "s"(S1), "s"(Z), "s"(Z)
                 : "memory");
    __builtin_amdgcn_s_wait_tensorcnt(0);           // drain before end
}

// ---------------------------------------------------------------------------
// Kernel 3: remainder rows via CDNA5 async Global<->LDS movers (ASYNCcnt).
// ---------------------------------------------------------------------------
#define AROWS 6   // TDM_TAIL == 6

__global__ __launch_bounds__(256) void gather_async_kernel(
    const float* __restrict__ vf,      // [B, N, D]
    const int*   __restrict__ ws_idx,  // [B, K]
    float*       __restrict__ out_sel, // [B, K, D]
    int j_base)
{
    __shared__ float4 buf[AROWS * 256]; // 24 KB LDS staging

    const int b  = blockIdx.y;
    const int j0 = j_base + blockIdx.x * AROWS;
    const int t  = threadIdx.x;

    unsigned lds_base = (unsigned)(uintptr_t)(void*)&buf[0];

    #pragma unroll
    for (int r = 0; r < AROWS; ++r) {
        int j   = j0 + r;
        int row = ws_idx[b * K_SEL + j];
        const float4* src =
            (const float4*)(vf + ((size_t)b * N_TOK + row) * D_FEAT) + t;
        unsigned lds = lds_base + (unsigned)((r * 256 + t) * sizeof(float4));
        unsigned long long ga = (unsigned long long)(uintptr_t)src;
        asm volatile("global_load_async_to_lds_b128 %0, %1, off"
                     :: "v"(lds), "v"(ga) : "memory");
    }
    asm volatile("s_wait_asynccnt 0x0" ::: "memory");

    #pragma unroll
    for (int r = 0; r < AROWS; ++r) {
        int j = j0 + r;
        float4* dst =
            (float4*)(out_sel + ((size_t)b * K_SEL + j) * D_FEAT) + t;
        unsigned lds = lds_base + (unsigned)((r * 256 + t) * sizeof(float4));
        unsigned long long ga = (unsigned long long)(uintptr_t)dst;
        asm volatile("global_store_async_from_lds_b128 %0, %1, off"
                     :: "v"(ga), "v"(lds) : "memory");
    }
    asm volatile("s_wait_asynccnt 0x0" ::: "memory");
}

// ---------------------------------------------------------------------------
extern "C" void kernel_launch(void* const* d_in, const int* in_sizes, int n_in,
                              void* d_out, int out_size, void* d_ws, size_t ws_size,
                              hipStream_t stream) {
    const float* vf     = (const float*)d_in[0];  // [32, 4096, 1024] f32
    const float* scores = (const float*)d_in[1];  // [32, 4096, 1]    f32

    float* out      = (float*)d_out;
    float* out_sel  = out + OUT_SEL_OFF;
    float* out_mask = out + OUT_MASK_OFF;
    float* out_idxf = out + OUT_IDX_OFF;

    int* ws_idx = (int*)d_ws;                     // B*K ints = ~205 KB

    // 1) top-k per batch: 32 blocks x 1024 threads (32 wave32s), 32 KB LDS.
    topk_sort_kernel<<<dim3(B_BATCH), dim3(1024), 0, stream>>>(
        scores, out_idxf, out_mask, ws_idx);

    // 2) TDM gather: 16 rows per wave-block, 64 KB LDS tile (rows 0..1631).
    gather_tdm_kernel<<<dim3(TDM_GROUPS, B_BATCH), dim3(32),
                        TDM_ROWS * D_FEAT * sizeof(float), stream>>>(
        vf, ws_idx, out_sel);

    // 3) async-LDS gather for the 6 remainder rows per batch (1632..1637).
    gather_async_kernel<<<dim3(1, B_BATCH), dim3(256), 0, stream>>>(
        vf, ws_idx, out_sel, TDM_GROUPS * TDM_ROWS);
}